// Flow_loss_68401649156835
// MI455X (gfx1250) — compile-verified
//
#include <hip/hip_runtime.h>
#include <cmath>

typedef float v2f __attribute__((ext_vector_type(2)));
typedef float v8f __attribute__((ext_vector_type(8)));

#define TPB 256
#define NWAVES (TPB / 32)

// ---------------------------------------------------------------------------
// Wave32 sum via the matrix pipe: D = A(16x4) * ones(4x16) + 0.
// Each lane contributes its partial as an A element; with B all-ones every
// column of D holds the 16 row-sums; column 0 lives in lane 0 (rows 0-7) and
// lane 16 (rows 8-15) of the f32 16x16 C/D layout. So:
//   wave_total = sum_r D[r] @lane0  +  sum_r D[r] @lane16
// 1 WMMA + 7 adds + 2 readlanes — no LDS traffic, runs on the matrix pipe.
// Requires EXEC all ones: callers must not diverge around this call.
// ---------------------------------------------------------------------------
__device__ __forceinline__ float wave_sum_wmma(float p) {
    v2f a; a[0] = p;   a[1] = 0.0f;
    v2f b; b[0] = 1.0f; b[1] = 1.0f;
    v8f c = {0.f, 0.f, 0.f, 0.f, 0.f, 0.f, 0.f, 0.f};
    v8f d = __builtin_amdgcn_wmma_f32_16x16x4_f32(
        /*neg_a=*/false, a, /*neg_b=*/false, b,
        /*c_mod=*/(short)0, c, /*reuse_a=*/false, /*reuse_b=*/false);
    float v = ((d[0] + d[1]) + (d[2] + d[3])) + ((d[4] + d[5]) + (d[6] + d[7]));
    int vi = __builtin_bit_cast(int, v);
    float r0 = __builtin_bit_cast(float, __builtin_amdgcn_readlane(vi, 0));
    float r1 = __builtin_bit_cast(float, __builtin_amdgcn_readlane(vi, 16));
    return r0 + r1;
}

// Block reduction: fixed TPB=256 -> 8 waves; tree-summed by thread 0.
__device__ __forceinline__ void block_atomic_add(float v, float* target) {
    __shared__ float lds[NWAVES];
    float w = wave_sum_wmma(v);
    unsigned wid = threadIdx.x >> 5;
    if ((threadIdx.x & 31u) == 0u) lds[wid] = w;
    __syncthreads();
    if (threadIdx.x == 0) {
        float s = ((lds[0] + lds[1]) + (lds[2] + lds[3])) +
                  ((lds[4] + lds[5]) + (lds[6] + lds[7]));
        atomicAdd(target, s);
    }
}

// ---------------------------------------------------------------------------
// Occlusion mask at one pixel: warp(ones, flow) = sum of valid bilinear
// weights (no gathers needed).
// ---------------------------------------------------------------------------
__device__ __forceinline__ float occ_mask_px(const float* __restrict__ flow,
                                             int p, int x, int y, int H, int W) {
    float fx = flow[2 * p], fy = flow[2 * p + 1];
    float xx = (float)x + fx, yy = (float)y + fy;
    float x0 = floorf(xx), y0 = floorf(yy);
    float m = 0.f;
#pragma unroll
    for (int dy = 0; dy < 2; ++dy) {
#pragma unroll
        for (int dx = 0; dx < 2; ++dx) {
            float xi = x0 + (float)dx, yi = y0 + (float)dy;
            float w  = (1.f - fabsf(xx - xi)) * (1.f - fabsf(yy - yi));
            bool valid = (xi >= 0.f) && (xi <= (float)(W - 1)) &&
                         (yi >= 0.f) && (yi <= (float)(H - 1));
            m += valid ? w : 0.f;
        }
    }
    return fminf(fmaxf(m, 0.f), 1.f);
}

// ---------------------------------------------------------------------------
// Slice (B,H,W,9) -> ct(3:6), s0(0:3), s1(6:9)
// ---------------------------------------------------------------------------
__global__ void slice_kernel(const float* __restrict__ in,
                             float* __restrict__ ct, float* __restrict__ s0,
                             float* __restrict__ s1, int npix) {
    int p = blockIdx.x * blockDim.x + threadIdx.x;
    if (p >= npix) return;
    const float* b = in + (size_t)p * 9;
    float* c = ct + (size_t)p * 3;
    float* a = s0 + (size_t)p * 3;
    float* d = s1 + (size_t)p * 3;
    a[0] = b[0]; a[1] = b[1]; a[2] = b[2];
    c[0] = b[3]; c[1] = b[4]; c[2] = b[5];
    d[0] = b[6]; d[1] = b[7]; d[2] = b[8];
}

// ---------------------------------------------------------------------------
// Factor-2 antialiased bilinear (triangle) downsample: [1,3,3,1]/8 per axis,
// edge taps renormalized per axis (jax.image.resize-style).
// ---------------------------------------------------------------------------
__global__ void down_kernel(const float* __restrict__ src, float* __restrict__ dst,
                            int Bn, int Hs, int Ws, int Hd, int Wd) {
    int idx = blockIdx.x * blockDim.x + threadIdx.x;
    int n = Bn * Hd * Wd * 3;
    if (idx >= n) return;
    int c = idx % 3; int t = idx / 3;
    int x = t % Wd; t /= Wd;
    int y = t % Hd; int b = t / Hd;
    const float wt[4] = {1.f, 3.f, 3.f, 1.f};
    float acc = 0.f, wys = 0.f;
#pragma unroll
    for (int j = 0; j < 4; ++j) {
        int ys = 2 * y - 1 + j;
        if (ys < 0 || ys >= Hs) continue;
        float rowacc = 0.f, wxs = 0.f;
#pragma unroll
        for (int i = 0; i < 4; ++i) {
            int xs = 2 * x - 1 + i;
            if (xs < 0 || xs >= Ws) continue;
            wxs += wt[i];
            rowacc += wt[i] * src[((size_t)(b * Hs + ys) * Ws + xs) * 3 + c];
        }
        acc += wt[j] * (rowacc / wxs);
        wys += wt[j];
    }
    dst[idx] = acc / wys;
}

// ---------------------------------------------------------------------------
// Store mask to buffer and accumulate its mean (ab) via WMMA reduction.
// ---------------------------------------------------------------------------
__global__ void mask_store_kernel(const float* __restrict__ flow,
                                  float* __restrict__ maskbuf,
                                  float* __restrict__ abp,
                                  int Bn, int H, int W, float invPix) {
    int n = Bn * H * W;
    int p = blockIdx.x * blockDim.x + threadIdx.x;
    float v = 0.f;
    if (p < n) {
        int x = p % W;
        int y = (p / W) % H;
        float m = occ_mask_px(flow, p, x, y, H, W);
        maskbuf[p] = m;
        v = m * invPix;
    }
    block_atomic_add(v, abp);
}

// ---------------------------------------------------------------------------
// Bilinear warp with validity masking (matches jnp gather-with-clip).
// ---------------------------------------------------------------------------
__global__ void warp_kernel(const float* __restrict__ img,
                            const float* __restrict__ flow,
                            float* __restrict__ out,
                            int Bn, int H, int W) {
    int n = Bn * H * W;
    int p = blockIdx.x * blockDim.x + threadIdx.x;
    if (p >= n) return;
    int x = p % W;
    int y = (p / W) % H;
    int b = p / (W * H);
    float fx = flow[2 * p], fy = flow[2 * p + 1];
    float xx = (float)x + fx, yy = (float)y + fy;
    float x0 = floorf(xx), y0 = floorf(yy);
    float o0 = 0.f, o1 = 0.f, o2 = 0.f;
#pragma unroll
    for (int dy = 0; dy < 2; ++dy) {
#pragma unroll
        for (int dx = 0; dx < 2; ++dx) {
            float xi = x0 + (float)dx, yi = y0 + (float)dy;
            float w  = (1.f - fabsf(xx - xi)) * (1.f - fabsf(yy - yi));
            bool valid = (xi >= 0.f) && (xi <= (float)(W - 1)) &&
                         (yi >= 0.f) && (yi <= (float)(H - 1));
            if (valid) {
                int xc = (int)xi, yc = (int)yi;
                xc = min(max(xc, 0), W - 1);
                yc = min(max(yc, 0), H - 1);
                const float* v = img + ((size_t)(b * H + yc) * W + xc) * 3;
                o0 += w * v[0]; o1 += w * v[1]; o2 += w * v[2];
            }
        }
    }
    float* o = out + (size_t)p * 3;
    o[0] = o0; o[1] = o1; o[2] = o2;
}

// ---------------------------------------------------------------------------
// recon term: mean(|p - ref| * m) / ab  (mean over B*H*W*3)
// ---------------------------------------------------------------------------
__global__ void recon_kernel(const float* __restrict__ pimg,
                             const float* __restrict__ ref,
                             const float* __restrict__ mask,
                             const float* __restrict__ abp,
                             float* __restrict__ acc,
                             int n, float invCnt) {
    int p = blockIdx.x * blockDim.x + threadIdx.x;
    float v = 0.f;
    if (p < n) {
        float m = mask[p];
        const float* a = pimg + (size_t)p * 3;
        const float* r = ref + (size_t)p * 3;
        float d = fabsf(a[0] - r[0]) + fabsf(a[1] - r[1]) + fabsf(a[2] - r[2]);
        float ab = *abp;
        v = d * m * invCnt / ab;
    }
    block_atomic_add(v, acc);
}

// ---------------------------------------------------------------------------
// ssim term: mean(ssim(p*m, ref*m)) / ab   (VALID 3x3 pooling, H-2 x W-2)
// ---------------------------------------------------------------------------
__global__ void ssim_kernel(const float* __restrict__ pimg,
                            const float* __restrict__ ref,
                            const float* __restrict__ mask,
                            const float* __restrict__ abp,
                            float* __restrict__ acc,
                            int Bn, int H, int W, float invCnt) {
    const float C1 = 0.0001f, C2 = 0.0009f;
    int Ho = H - 2, Wo = W - 2;
    int n = Bn * Ho * Wo;
    int idx = blockIdx.x * blockDim.x + threadIdx.x;
    float v = 0.f;
    if (idx < n) {
        int ox = idx % Wo;
        int oy = (idx / Wo) % Ho;
        int b  = idx / (Wo * Ho);
        // Prefetch the row below the 3x3 window (next output row's new data)
        // into the WGP cache: gfx1250 global_prefetch path.
        if (oy + 3 < H) {
            int pnext = (b * H + (oy + 3)) * W + ox;
            __builtin_prefetch(pimg + (size_t)pnext * 3, 0, 1);
            __builtin_prefetch(ref + (size_t)pnext * 3, 0, 1);
        }
        float sx[3] = {0.f, 0.f, 0.f}, sy[3] = {0.f, 0.f, 0.f};
        float sxx[3] = {0.f, 0.f, 0.f}, syy[3] = {0.f, 0.f, 0.f}, sxy[3] = {0.f, 0.f, 0.f};
#pragma unroll
        for (int j = 0; j < 3; ++j) {
#pragma unroll
            for (int i = 0; i < 3; ++i) {
                int pp = (b * H + (oy + j)) * W + (ox + i);
                float m = mask[pp];
                const float* a = pimg + (size_t)pp * 3;
                const float* r = ref + (size_t)pp * 3;
#pragma unroll
                for (int c = 0; c < 3; ++c) {
                    float xv = a[c] * m, yv = r[c] * m;
                    sx[c] += xv; sy[c] += yv;
                    sxx[c] += xv * xv; syy[c] += yv * yv; sxy[c] += xv * yv;
                }
            }
        }
        const float inv9 = 1.f / 9.f;
        float tot = 0.f;
#pragma unroll
        for (int c = 0; c < 3; ++c) {
            float mx = sx[c] * inv9, my = sy[c] * inv9;
            float vx = sxx[c] * inv9 - mx * mx;
            float vy = syy[c] * inv9 - my * my;
            float vxy = sxy[c] * inv9 - mx * my;
            float nume = (2.f * mx * my + C1) * (2.f * vxy + C2);
            float deno = (mx * mx + my * my + C1) * (vx + vy + C2);
            float val = (1.f - nume / deno) * 0.5f;
            tot += fminf(fmaxf(val, 0.f), 1.f);
        }
        float ab = *abp;
        v = tot * invCnt / ab;
    }
    block_atomic_add(v, acc);
}

// ---------------------------------------------------------------------------
// 2nd-order edge-aware smoothness on flow/20.
// ---------------------------------------------------------------------------
__global__ void smooth_kernel(const float* __restrict__ flow,
                              const float* __restrict__ img,
                              float* __restrict__ acc,
                              int Bn, int H, int W, float invX, float invY) {
    int n = Bn * H * W;
    int idx = blockIdx.x * blockDim.x + threadIdx.x;
    float v = 0.f;
    if (idx < n) {
        int x = idx % W;
        int y = (idx / W) % H;
        int b = idx / (W * H);
        int base = (b * H + y) * W + x;
        if (x <= W - 3) {
            const float* i1 = img + (size_t)(base + 1) * 3;
            const float* i2 = img + (size_t)(base + 2) * 3;
            float g = fabsf(i2[0] - i1[0]) + fabsf(i2[1] - i1[1]) + fabsf(i2[2] - i1[2]);
            float wx = expf(-10.f * g * (1.f / 3.f));
            const float* f0 = flow + (size_t)base * 2;
            const float* f1 = flow + (size_t)(base + 1) * 2;
            const float* f2 = flow + (size_t)(base + 2) * 2;
            float a0 = (f2[0] - 2.f * f1[0] + f0[0]) * 0.05f;
            float a1 = (f2[1] - 2.f * f1[1] + f0[1]) * 0.05f;
            v += wx * (fabsf(a0) + fabsf(a1)) * invX;
        }
        if (y <= H - 3) {
            const float* i1 = img + (size_t)(base + W) * 3;
            const float* i2 = img + (size_t)(base + 2 * W) * 3;
            float g = fabsf(i2[0] - i1[0]) + fabsf(i2[1] - i1[1]) + fabsf(i2[2] - i1[2]);
            float wy = expf(-10.f * g * (1.f / 3.f));
            const float* f0 = flow + (size_t)base * 2;
            const float* f1 = flow + (size_t)(base + W) * 2;
            const float* f2 = flow + (size_t)(base + 2 * W) * 2;
            float a0 = (f2[0] - 2.f * f1[0] + f0[0]) * 0.05f;
            float a1 = (f2[1] - 2.f * f1[1] + f0[1]) * 0.05f;
            v += wy * (fabsf(a0) + fabsf(a1)) * invY;
        }
    }
    block_atomic_add(v, acc);
}

__global__ void final_kernel(const float* __restrict__ accs, float* __restrict__ out) {
    const float SSIM_W = 0.85f, RECON_W = 0.15f, SMOOTH_W = 10.0f, CROSS_W = 0.3f;
    if (blockIdx.x == 0 && threadIdx.x == 0) {
        float recon = accs[24], crecon = accs[25];
        float ssim  = accs[26], cssim  = accs[27];
        float smoo  = accs[28], csmoo  = accs[29];
        out[0] = RECON_W * (recon + CROSS_W * crecon) +
                 SSIM_W  * (ssim  + CROSS_W * cssim)  +
                 SMOOTH_W * (smoo + CROSS_W * csmoo);
    }
}

// ---------------------------------------------------------------------------
// Host orchestration
// ---------------------------------------------------------------------------
static inline int grid_for(int n) { return (n + TPB - 1) / TPB; }

extern "C" void kernel_launch(void* const* d_in, const int* in_sizes, int n_in,
                              void* d_out, int out_size, void* d_ws, size_t ws_size,
                              hipStream_t stream) {
    (void)in_sizes; (void)n_in; (void)out_size; (void)ws_size;
    const int Bn = 8, H0c = 256, W0c = 832;

    const float* in_img = (const float*)d_in[0];
    // flow id f = dir*3 + i   (fw0,fw1,fw2,bw0,bw1,bw2); d_in index = 1 + f*4 + s
    auto flowPtr = [&](int f, int s) -> const float* {
        return (const float*)d_in[1 + f * 4 + s];
    };

    float* ws = (float*)d_ws;
    float* accs = ws; // [0..23]=ab per (flow,scale); 24..29 = loss accumulators
    hipMemsetAsync(accs, 0, 64 * sizeof(float), stream);

    int Hs[4], Wss[4]; int Ns[4];
    for (int s = 0; s < 4; ++s) { Hs[s] = H0c >> s; Wss[s] = W0c >> s; Ns[s] = Bn * Hs[s] * Wss[s]; }

    size_t off = 64;
    float* img[4][3]; // [scale][0=ct,1=s0,2=s1]
    for (int s = 0; s < 4; ++s)
        for (int k = 0; k < 3; ++k) { img[s][k] = ws + off; off += (size_t)Ns[s] * 3; }
    float* warpbuf = ws + off; off += (size_t)Ns[0] * 3;
    float* maskbuf = ws + off; off += (size_t)Ns[0];

    // 1) slice full-res channels
    slice_kernel<<<grid_for(Ns[0]), TPB, 0, stream>>>(in_img, img[0][0], img[0][1], img[0][2], Ns[0]);

    // 2) pyramid (iterated antialiased 2x downsample)
    for (int s = 1; s < 4; ++s)
        for (int k = 0; k < 3; ++k) {
            int n = Ns[s] * 3;
            down_kernel<<<grid_for(n), TPB, 0, stream>>>(img[s - 1][k], img[s][k],
                                                         Bn, Hs[s - 1], Wss[s - 1], Hs[s], Wss[s]);
        }

    // pair table: {warp-src img, ref img, warp-flow f, mask-flow f, cross}
    struct PairDesc { int src, ref, wf, mf, cross; };
    const PairDesc pairs[6] = {
        {0, 1, 0, 3, 0},  // p_src0   = warp(ct, fw0) vs s0, mask bw0
        {2, 1, 2, 5, 1},  // p_src0_1 = warp(s1, fw2) vs s0, mask bw2   (cross)
        {2, 0, 1, 4, 0},  // p_tgt    = warp(s1, fw1) vs ct, mask bw1
        {1, 0, 3, 0, 0},  // p_tgt_1  = warp(s0, bw0) vs ct, mask fw0
        {0, 2, 4, 1, 0},  // p_src1   = warp(ct, bw1) vs s1, mask fw1
        {1, 2, 5, 2, 1},  // p_src1_1 = warp(s0, bw2) vs s1, mask fw2   (cross)
    };
    // smoothness table: {flow f, img, cross}
    const int sm[6][3] = { {0,1,0}, {1,0,0}, {2,1,1}, {3,0,0}, {4,2,0}, {5,2,1} };

    for (int s = 0; s < 4; ++s) {
        int H = Hs[s], W = Wss[s], n = Ns[s];
        float invPix = 1.0f / (float)n;
        float invRecon = 1.0f / (float)((size_t)n * 3);
        float invSsim = 1.0f / (float)((size_t)Bn * (H - 2) * (W - 2) * 3);

        for (int t = 0; t < 6; ++t) {
            const PairDesc& P = pairs[t];
            const float* mfl = flowPtr(P.mf, s);
            const float* wfl = flowPtr(P.wf, s);
            float* abp = accs + (P.mf * 4 + s);
            float* racc = accs + (P.cross ? 25 : 24);
            float* sacc = accs + (P.cross ? 27 : 26);

            mask_store_kernel<<<grid_for(n), TPB, 0, stream>>>(mfl, maskbuf, abp, Bn, H, W, invPix);
            warp_kernel<<<grid_for(n), TPB, 0, stream>>>(img[s][P.src], wfl, warpbuf, Bn, H, W);
            recon_kernel<<<grid_for(n), TPB, 0, stream>>>(warpbuf, img[s][P.ref], maskbuf, abp,
                                                          racc, n, invRecon);
            int nss = Bn * (H - 2) * (W - 2);
            ssim_kernel<<<grid_for(nss), TPB, 0, stream>>>(warpbuf, img[s][P.ref], maskbuf, abp,
                                                           sacc, Bn, H, W, invSsim);
        }

        float invX = 0.5f / (float)((size_t)Bn * H * (W - 2) * 2);
        float invY = 0.5f / (float)((size_t)Bn * (H - 2) * W * 2);
        for (int t = 0; t < 6; ++t) {
            float* macc = accs + (sm[t][2] ? 29 : 28);
            smooth_kernel<<<grid_for(n), TPB, 0, stream>>>(flowPtr(sm[t][0], s), img[s][sm[t][1]],
                                                           macc, Bn, H, W, invX, invY);
        }
    }

    final_kernel<<<1, 64, 0, stream>>>(accs, (float*)d_out);
}